// MultiplierLUTONNX_35038343201110
// MI455X (gfx1250) — compile-verified
//
#include <hip/hip_runtime.h>
#include <cstdint>
#include <cstddef>

// MultiplierLUTONNX: out[b, :] = sigmoid(table[a_idx[b], b_idx[b], :])
// a_idx/b_idx recovered from true one-hot f32 rows.
//
// Roofline: 2 x 64MB one-hot streams + ~4MB L2-resident table gather + 4MB out
// => ~136MB => ~5.8us at 23.3 TB/s.  Data path: gfx1250 async global->LDS
// engine (ASYNCcnt), double buffered.  Index recovery is a real matrix op:
//   idx[m] = onehot[m,:] . iota[256]
// batched 16 rows (8 a-rows + 8 b-rows) per wave as a chain of 64
// V_WMMA_F32_16X16X4_F32 ops with a VALU-synthesized broadcast-iota B operand.
// Products are 1.0*k -> exact in f32.

typedef float v2f __attribute__((ext_vector_type(2)));
typedef float v8f __attribute__((ext_vector_type(8)));

#define N_ONEHOT    256   // one-hot width (floats per row)
#define KOUT        16    // output floats per row
#define WAVES_PB    8     // waves per block (256 threads, wave32)
#define ROWS_PT     8     // batch rows per tile (A rows 0-7 = a, 8-15 = b)
#define LDS_STRIDE  260   // padded floats per row: bank advance 4/row -> conflict-free
#define TILE_GBYTES 8192  // 8 rows * 1KB per input

// Async 16B-per-lane copy: copies 16 bytes from the global address sbase + voff to the LDS offset lds_off. GVS addressing.
__device__ __forceinline__ void async_b128(uint32_t lds_off, uint64_t sbase, uint32_t voff) {
    asm volatile("global_load_async_to_lds_b128 %0, %1, %2"
                 :: "v"(lds_off), "v"(voff), "s"(sbase)
                 : "memory");
}

__global__ __launch_bounds__(256)
void onehot_lut_sigmoid_kernel(const float* __restrict__ a1h,
                               const float* __restrict__ b1h,
                               const float* __restrict__ table,
                               float* __restrict__ out,
                               int B) {
    // [wave][buf][16 padded rows]: 8 * 2 * 16*260 * 4B = 266,240 B (<320KB WG max)
    __shared__ float smem[WAVES_PB][2][16 * LDS_STRIDE];

    const int tid    = threadIdx.x;
    const int wave   = tid >> 5;
    const int lane   = tid & 31;
    const int gwave  = blockIdx.x * WAVES_PB + wave;
    const int nwaves = gridDim.x * WAVES_PB;

    const int m    = lane & 15;   // A-matrix row owned by this lane
    const int half = lane >> 4;   // K sub-pair selector (K = i0 + half*2 + {0,1})

    const uint32_t lane16 = (uint32_t)lane * 16u;
    const uint64_t abase  = (uint64_t)(uintptr_t)a1h;
    const uint64_t bbase  = (uint64_t)(uintptr_t)b1h;

    // Low 32 bits of the flat shared-aperture address == LDS byte offset.
    const uint32_t ldsBuf[2] = {
        (uint32_t)(uintptr_t)&smem[wave][0][0],
        (uint32_t)(uintptr_t)&smem[wave][1][0]
    };

    // Stage tile t: 8 a-rows into padded rows 0-7, 8 b-rows into rows 8-15.
    // 32 async b128 ops (512B contiguous per op on the global side).
    auto issue = [&](int bi_, int t) {
        const uint32_t g0 = (uint32_t)t * TILE_GBYTES + lane16;
        const uint32_t l0 = ldsBuf[bi_] + lane16;
        #pragma unroll
        for (int j = 0; j < ROWS_PT; ++j) {
            const uint32_t la = l0 + (uint32_t)(j * (LDS_STRIDE * 4));
            const uint32_t lb = la + (uint32_t)(8 * (LDS_STRIDE * 4));
            const uint32_t g  = g0 + (uint32_t)(j * 1024);
            async_b128(la,        abase, g);
            async_b128(la + 512u, abase, g + 512u);
            async_b128(lb,        bbase, g);
            async_b128(lb + 512u, bbase, g + 512u);
        }
    };

    const int ntiles = B / ROWS_PT;
    int t  = gwave;
    int bi = 0;
    if (t < ntiles) issue(bi, t);

    for (; t < ntiles; t += nwaves) {
        const int tn = t + nwaves;
        if (tn < ntiles) {
            issue(bi ^ 1, tn);                                  // prefetch next tile
            asm volatile("s_wait_asynccnt 0x20" ::: "memory");  // retire current 32
        } else {
            asm volatile("s_wait_asynccnt 0x0" ::: "memory");
        }

        // ---- WMMA iota-dot: D[16x16] = A[16x256] x IotaChunks, 64 x K=4 steps ----
        // A frag: lane m holds row m, cols i0 + half*2 + {0,1}  (ds_load_b64)
        // B frag: B[k,n] = i0 + k  -> per lane {i0+half*2, i0+half*2+1}
        const float* rowp = &smem[wave][bi][0] + m * LDS_STRIDE + half * 2;
        const float  cbase = (float)(half * 2);
        v8f acc = {};
        #pragma unroll
        for (int i0 = 0; i0 < N_ONEHOT; i0 += 4) {
            const v2f a = *(const v2f*)(rowp + i0);
            v2f b;
            b.x = cbase + (float)i0;
            b.y = b.x + 1.0f;
            acc = __builtin_amdgcn_wmma_f32_16x16x4_f32(
                /*neg_a=*/false, a, /*neg_b=*/false, b,
                /*c_mod=*/(short)0, acc, /*reuse_a=*/false, /*reuse_b=*/false);
        }
        // acc[j]: lanes 0-15 = a_idx(row j); lanes 16-31 = b_idx(row j). Exact ints.

        const size_t r0 = (size_t)t * ROWS_PT;
        #pragma unroll
        for (int j = 0; j < ROWS_PT; ++j) {
            const float iaf = __shfl(acc[j], 0, 32);
            const float ibf = __shfl(acc[j], 16, 32);
            const int ia = (int)(iaf + 0.5f);
            const int ib = (int)(ibf + 0.5f);
            if (lane < KOUT) {
                const float tv = table[(((size_t)ia << 8) + (size_t)ib) * KOUT + lane];
                out[(r0 + j) * KOUT + lane] = 1.0f / (1.0f + __expf(-tv));
            }
        }

        bi ^= 1;
    }
}

extern "C" void kernel_launch(void* const* d_in, const int* in_sizes, int n_in,
                              void* d_out, int out_size, void* d_ws, size_t ws_size,
                              hipStream_t stream) {
    const float* a1h   = (const float*)d_in[0]; // [B, 256] f32 one-hot
    const float* b1h   = (const float*)d_in[1]; // [B, 256] f32 one-hot
    const float* table = (const float*)d_in[2]; // [256, 256, 16] f32
    float*       out   = (float*)d_out;         // [B, 16] f32

    const int B = in_sizes[0] / N_ONEHOT;
    const int ntiles = B / ROWS_PT;             // B = 65536 -> 8192 tiles

    // 512 blocks x 8 waves = 4096 waves -> 2 tiles per wave (double buffer used).
    int blocks = 512;
    const int maxBlocks = (ntiles + WAVES_PB - 1) / WAVES_PB;
    if (blocks > maxBlocks) blocks = maxBlocks;
    if (blocks < 1) blocks = 1;

    onehot_lut_sigmoid_kernel<<<blocks, 256, 0, stream>>>(a1h, b1h, table, out, B);
}